// GNNModel_26345329394307
// MI455X (gfx1250) — compile-verified
//
#include <hip/hip_runtime.h>

#define FD 128        // feature dim
#define LDSPITCH 132  // padded LDS row pitch (floats): lanes map to distinct banks

typedef __attribute__((ext_vector_type(2))) float v2f;
typedef __attribute__((ext_vector_type(8))) float v8f;

// ---------------------------------------------------------------------------
// Zero a float buffer (n4 = count of float4)
// ---------------------------------------------------------------------------
__global__ void zero_f32x4_kernel(float4* __restrict__ p, long n4) {
    long i = (long)blockIdx.x * blockDim.x + threadIdx.x;
    if (i < n4) p[i] = make_float4(0.f, 0.f, 0.f, 0.f);
}

// ---------------------------------------------------------------------------
// Per-destination edge counts (mean denominator). One thread per edge.
// ---------------------------------------------------------------------------
__global__ void edge_count_kernel(const int* __restrict__ edge, float* __restrict__ cnt, int E) {
    int e = blockIdx.x * blockDim.x + threadIdx.x;
    if (e >= E) return;
    int dst = edge[E + e];
    unsafeAtomicAdd(cnt + dst, 1.0f);   // global_atomic_add_f32
}

// ---------------------------------------------------------------------------
// Scatter-add of gathered source rows: agg[dst] += x_src[src].
// 32 threads per edge, float4 (16B) per thread -> 128 floats per edge.
// x_src (51.2MB) fits in the 192MB L2, so gathers are mostly L2 hits.
// ---------------------------------------------------------------------------
__global__ void scatter_add_kernel(const float* __restrict__ xsrc,
                                   const int* __restrict__ edge,
                                   float* __restrict__ agg, int E) {
    long t = (long)blockIdx.x * blockDim.x + threadIdx.x;
    int e = (int)(t >> 5);
    if (e >= E) return;
    int c = (int)(t & 31);
    int src = edge[e];
    int dst = edge[E + e];
    float4 v = reinterpret_cast<const float4*>(xsrc + (size_t)src * FD)[c];
    float* p = agg + (size_t)dst * FD + c * 4;
    unsafeAtomicAdd(p + 0, v.x);
    unsafeAtomicAdd(p + 1, v.y);
    unsafeAtomicAdd(p + 2, v.z);
    unsafeAtomicAdd(p + 3, v.w);
}

// ---------------------------------------------------------------------------
// Fused SAGE dense part:
//   out = relu?( (agg / max(cnt,1)) @ Wl^T + bl + xdst @ Wr^T )
// Block = 256 threads = 8 waves; block covers 16 rows x 128 output channels,
// one 16x16 tile per wave via V_WMMA_F32_16X16X4_F32 (keeps fp32 numerics;
// workload is memory-bound so low-precision buys nothing).
//
// The 16x128 A tiles (agg + xdst) are staged once per block into padded LDS
// using async DMA (global_load_async_to_lds_b128, ASYNCcnt), then WMMA A
// fragments come from conflict-free ds_load_b64:
//   lane r reads sTile[r*132 + k0 + 2*half] -> banks {4r, 4r+2}, all distinct.
// B (weights) stays on direct global loads: the same 16x128 W slices are
// re-read by every block on a WGP -> L0/L2 resident.
// ---------------------------------------------------------------------------
__global__ __launch_bounds__(256) void sage_gemm_wmma_kernel(
    const float* __restrict__ agg, const float* __restrict__ cnt,
    const float* __restrict__ xdst,
    const float* __restrict__ Wl, const float* __restrict__ bl,
    const float* __restrict__ Wr,
    float* __restrict__ out, int nrows, int relu)
{
    __shared__ __attribute__((aligned(16))) float sAgg[16 * LDSPITCH];
    __shared__ __attribute__((aligned(16))) float sX[16 * LDSPITCH];

    const int tid  = threadIdx.x;
    const int wave = tid >> 5;
    const int lane = tid & 31;
    const int half = lane >> 4;     // 0: K pair {0,1}, 1: K pair {2,3}
    const int r    = lane & 15;
    const int row0 = blockIdx.x * 16;
    const int o0   = wave * 16;     // output-channel tile
    if (row0 >= nrows) return;      // block-uniform; EXEC stays all-ones

    // ---- async-DMA stage both 16x128 A tiles into padded LDS -------------
    {
        const unsigned ldsAgg = (unsigned)(size_t)&sAgg[0];
        const unsigned ldsX   = (unsigned)(size_t)&sX[0];
#pragma unroll
        for (int s = 0; s < 2; ++s) {
            int c   = tid + s * 256;      // 512 x 16B chunks per tile
            int row = c >> 5;
            int col = c & 31;             // 16B column chunk
            unsigned loff = (unsigned)(row * LDSPITCH + col * 4) * 4u;
            unsigned long long ga =
                (unsigned long long)(agg  + (size_t)(row0 + row) * FD + col * 4);
            unsigned long long gx =
                (unsigned long long)(xdst + (size_t)(row0 + row) * FD + col * 4);
            unsigned la = ldsAgg + loff;
            unsigned lx = ldsX + loff;
            asm volatile("global_load_async_to_lds_b128 %0, %1, off"
                         :: "v"(la), "v"(ga) : "memory");
            asm volatile("global_load_async_to_lds_b128 %0, %1, off"
                         :: "v"(lx), "v"(gx) : "memory");
        }
        asm volatile("s_wait_asynccnt 0x0" ::: "memory");
    }
    __syncthreads();

    // ---- WMMA accumulation ----------------------------------------------
    // bias depends only on output channel -> same value in all 8 acc regs
    const float bias = bl[o0 + r];
    v8f acc;
#pragma unroll
    for (int j = 0; j < 8; ++j) acc[j] = bias;

    const float inv = 1.0f / fmaxf(cnt[row0 + r], 1.0f);

    const float* wlrow = Wl + (size_t)(o0 + r) * FD;
    const float* wrrow = Wr + (size_t)(o0 + r) * FD;
    const int koff = 2 * half;
    const float* sa = &sAgg[r * LDSPITCH + koff];
    const float* sx = &sX[r * LDSPITCH + koff];

    // term 1: mean-aggregated neighbors x Wl^T (normalization folded in)
#pragma unroll 4
    for (int k0 = 0; k0 < FD; k0 += 4) {
        v2f a = *reinterpret_cast<const v2f*>(sa + k0);
        a.x *= inv; a.y *= inv;
        v2f b = *reinterpret_cast<const v2f*>(wlrow + k0 + koff);
        acc = __builtin_amdgcn_wmma_f32_16x16x4_f32(
            false, a, false, b, (short)0, acc, false, false);
    }
    // term 2: self features x Wr^T
#pragma unroll 4
    for (int k0 = 0; k0 < FD; k0 += 4) {
        v2f a = *reinterpret_cast<const v2f*>(sx + k0);
        v2f b = *reinterpret_cast<const v2f*>(wrrow + k0 + koff);
        acc = __builtin_amdgcn_wmma_f32_16x16x4_f32(
            false, a, false, b, (short)0, acc, false, false);
    }

    if (relu) {
#pragma unroll
        for (int j = 0; j < 8; ++j) acc[j] = fmaxf(acc[j], 0.f);
    }
    // D layout: VGPR j = D[M = j + 8*half][N = lane&15]
#pragma unroll
    for (int j = 0; j < 8; ++j) {
        int row = row0 + j + 8 * half;
        out[(size_t)row * FD + o0 + r] = acc[j];
    }
}

// ---------------------------------------------------------------------------
// Host-side orchestration
// ---------------------------------------------------------------------------
static inline int cdiv(long a, int b) { return (int)((a + b - 1) / b); }

extern "C" void kernel_launch(void* const* d_in, const int* in_sizes, int n_in,
                              void* d_out, int out_size, void* d_ws, size_t ws_size,
                              hipStream_t stream) {
    const float* x_m = (const float*)d_in[0];
    const float* x_t = (const float*)d_in[1];
    const float* Wl  = (const float*)d_in[2];   // [3][2][128][128]
    const float* bl  = (const float*)d_in[3];   // [3][2][128]
    const float* Wr  = (const float*)d_in[4];   // [3][2][128][128]
    const int* e_mt  = (const int*)d_in[5];     // [2][E] src=row0, dst=row1
    const int* e_tm  = (const int*)d_in[6];

    const int NM   = in_sizes[0] / FD;
    const int NT   = in_sizes[1] / FD;
    const int E_mt = in_sizes[5] / 2;
    const int E_tm = in_sizes[6] / 2;

    float* out_m = (float*)d_out;
    float* out_t = out_m + (size_t)NM * FD;

    // workspace carve-up: agg + one ping-pong pair + counts (~155MB).
    // d_out doubles as the layer-0 output buffer (layer 2 overwrites it).
    const size_t featM = (size_t)NM * FD, featT = (size_t)NT * FD;
    const size_t featMax = featM > featT ? featM : featT;
    char* w = (char*)d_ws;
    float* agg   = (float*)w; w += featMax * sizeof(float);
    float* bufM  = (float*)w; w += featM * sizeof(float);
    float* bufT  = (float*)w; w += featT * sizeof(float);
    float* cnt_t = (float*)w; w += (size_t)((NT + 3) & ~3) * sizeof(float);
    float* cnt_m = (float*)w; w += (size_t)((NM + 3) & ~3) * sizeof(float);

    const int ZB = 256;

    // edge counts: depend only on edges -> once per launch
    zero_f32x4_kernel<<<cdiv((NT + 3) / 4, ZB), ZB, 0, stream>>>((float4*)cnt_t, (NT + 3) / 4);
    zero_f32x4_kernel<<<cdiv((NM + 3) / 4, ZB), ZB, 0, stream>>>((float4*)cnt_m, (NM + 3) / 4);
    edge_count_kernel<<<cdiv(E_mt, ZB), ZB, 0, stream>>>(e_mt, cnt_t, E_mt);
    edge_count_kernel<<<cdiv(E_tm, ZB), ZB, 0, stream>>>(e_tm, cnt_m, E_tm);

    const float* cm = x_m;
    const float* ct = x_t;
    for (int layer = 0; layer < 3; ++layer) {
        float* nm; float* nt;
        if (layer == 1) { nm = bufM;  nt = bufT;  }
        else            { nm = out_m; nt = out_t; }   // layers 0 and 2
        const int relu = (layer < 2) ? 1 : 0;

        const float* wl0 = Wl + ((size_t)layer * 2 + 0) * FD * FD;
        const float* wl1 = Wl + ((size_t)layer * 2 + 1) * FD * FD;
        const float* wr0 = Wr + ((size_t)layer * 2 + 0) * FD * FD;
        const float* wr1 = Wr + ((size_t)layer * 2 + 1) * FD * FD;
        const float* bl0 = bl + ((size_t)layer * 2 + 0) * FD;
        const float* bl1 = bl + ((size_t)layer * 2 + 1) * FD;

        // edge type 0: musician -> track (dst = track)
        zero_f32x4_kernel<<<cdiv(featT / 4, ZB), ZB, 0, stream>>>((float4*)agg, featT / 4);
        scatter_add_kernel<<<cdiv((long)E_mt * 32, ZB), ZB, 0, stream>>>(cm, e_mt, agg, E_mt);
        sage_gemm_wmma_kernel<<<cdiv(NT, 16), 256, 0, stream>>>(
            agg, cnt_t, ct, wl0, bl0, wr0, nt, NT, relu);

        // edge type 1: track -> musician (dst = musician)
        zero_f32x4_kernel<<<cdiv(featM / 4, ZB), ZB, 0, stream>>>((float4*)agg, featM / 4);
        scatter_add_kernel<<<cdiv((long)E_tm * 32, ZB), ZB, 0, stream>>>(ct, e_tm, agg, E_tm);
        sage_gemm_wmma_kernel<<<cdiv(NM, 16), 256, 0, stream>>>(
            agg, cnt_m, cm, wl1, bl1, wr1, nm, NM, relu);

        cm = nm; ct = nt;
    }
}